// ChamferLoss_76175539962211
// MI455X (gfx1250) — compile-verified
//
#include <hip/hip_runtime.h>
#include <hip/hip_bf16.h>

typedef __attribute__((ext_vector_type(2))) float v2f;
typedef __attribute__((ext_vector_type(8))) float v8f;

// ---------------------------------------------------------------------------
// Pass 0: per point i, write
//   - packed float4 (c0,c1,c2,|p|^2)  (used for A fragments + row bias)
//   - its two B-fragment slots in WMMA lane layout:
//       tile = i>>4, sub = i&15
//       lane sub      (K=0/1): {c0, c1}
//       lane 16+sub   (K=2/3): {c2, |p|^2}
//   - running-min init (+INF)
// ---------------------------------------------------------------------------
__global__ __launch_bounds__(256) void chamfer_pack_init(
    const float* __restrict__ Xc, const float* __restrict__ Xt,
    float4* __restrict__ Tc, float4* __restrict__ Tt,
    float2* __restrict__ BfC, float2* __restrict__ BfT,
    float* __restrict__ rmC, float* __restrict__ rmT, int N, int M) {
  int i = blockIdx.x * blockDim.x + threadIdx.x;
  if (i < N) {
    float x0 = Xc[3 * i + 0], x1 = Xc[3 * i + 1], x2 = Xc[3 * i + 2];
    float n = x0 * x0 + x1 * x1 + x2 * x2;
    Tc[i] = make_float4(x0, x1, x2, n);
    int tile = i >> 4, sub = i & 15;
    BfC[tile * 32 + sub]      = make_float2(x0, x1);
    BfC[tile * 32 + 16 + sub] = make_float2(x2, n);
    rmC[i] = __builtin_inff();
  }
  if (i < M) {
    float y0 = Xt[3 * i + 0], y1 = Xt[3 * i + 1], y2 = Xt[3 * i + 2];
    float n = y0 * y0 + y1 * y1 + y2 * y2;
    Tt[i] = make_float4(y0, y1, y2, n);
    int tile = i >> 4, sub = i & 15;
    BfT[tile * 32 + sub]      = make_float2(y0, y1);
    BfT[tile * 32 + 16 + sub] = make_float2(y2, n);
    rmT[i] = __builtin_inff();
  }
}

// ---------------------------------------------------------------------------
// Pass 1 (x2 directions): each wave owns 32 query rows (two 16-row A tiles),
// scans a chunk of target columns. Per column tile: one b64 fragment load
// feeds two V_WMMA_F32_16X16X4_F32 ops producing t_ij = |y_j|^2 - 2 x.y;
// running mins kept in two 8-VGPR D-layout accumulators.
// ---------------------------------------------------------------------------
__global__ __launch_bounds__(256) void chamfer_min_kernel(
    const float4* __restrict__ Q,   // packed query points (A rows + bias)
    const v2f*    __restrict__ Bf,  // pre-packed B fragments, 32 lanes/tile
    float* __restrict__ rm,         // per-row running min (uint-bit atomicMin)
    int tiles_per_chunk) {
  const int lane = threadIdx.x & 31;
  const int wave = threadIdx.x >> 5;
  const int i0   = (blockIdx.x * 8 + wave) * 32;  // first of 32 query rows
  const int half = lane >> 4;                     // 0: K=0/1, 1: K=2/3
  const int sub  = lane & 15;

  // Two A fragments: row i = [-2x0, -2x1, -2x2, 1]
  const float4 q0 = Q[i0 + sub];
  const float4 q1 = Q[i0 + 16 + sub];
  v2f a0, a1;
  a0.x = half ? (-2.0f * q0.z) : (-2.0f * q0.x);
  a0.y = half ? 1.0f           : (-2.0f * q0.y);
  a1.x = half ? (-2.0f * q1.z) : (-2.0f * q1.x);
  a1.y = half ? 1.0f           : (-2.0f * q1.y);

  v8f acc0, acc1;
#pragma unroll
  for (int k = 0; k < 8; ++k) { acc0[k] = __builtin_inff(); acc1[k] = __builtin_inff(); }

  const int tbase = blockIdx.y * tiles_per_chunk;
  const v2f* bp = Bf + (size_t)tbase * 32 + lane;
  for (int jt = 0; jt < tiles_per_chunk; ++jt) {
    const v2f b = bp[jt * 32];          // one b64 load, no lane shuffling
    v8f c0 = {}, c1 = {};
    v8f d0 = __builtin_amdgcn_wmma_f32_16x16x4_f32(
        false, a0, false, b, (short)0, c0, false, false);
    v8f d1 = __builtin_amdgcn_wmma_f32_16x16x4_f32(
        false, a1, false, b, (short)0, c1, false, false);
#pragma unroll
    for (int k = 0; k < 8; ++k) {
      acc0[k] = fminf(acc0[k], d0[k]);
      acc1[k] = fminf(acc1[k], d1[k]);
    }
  }

  // Reduce the 16 column-lanes within each half (xor masks < 16 stay in-half).
  float red0[8], red1[8];
#pragma unroll
  for (int k = 0; k < 8; ++k) {
    float v = acc0[k];
    v = fminf(v, __shfl_xor(v, 1, 32));
    v = fminf(v, __shfl_xor(v, 2, 32));
    v = fminf(v, __shfl_xor(v, 4, 32));
    v = fminf(v, __shfl_xor(v, 8, 32));
    red0[k] = v;
    float w = acc1[k];
    w = fminf(w, __shfl_xor(w, 1, 32));
    w = fminf(w, __shfl_xor(w, 2, 32));
    w = fminf(w, __shfl_xor(w, 4, 32));
    w = fminf(w, __shfl_xor(w, 8, 32));
    red1[k] = w;
  }

  // D layout: VGPR k holds row M=k (lanes 0-15) / M=k+8 (lanes 16-31).
  if (sub == 0) {
    const int rb0 = i0 + half * 8;
    const int rb1 = i0 + 16 + half * 8;
#pragma unroll
    for (int k = 0; k < 8; ++k) {
      // d2 = |x_i|^2 + min_j t_ij, clamped at 0 (clamp commutes with min).
      float v0 = fmaxf(red0[k] + Q[rb0 + k].w, 0.0f);
      atomicMin((unsigned int*)&rm[rb0 + k], __float_as_uint(v0));  // all >= 0
      float v1 = fmaxf(red1[k] + Q[rb1 + k].w, 0.0f);
      atomicMin((unsigned int*)&rm[rb1 + k], __float_as_uint(v1));
    }
  }
}

// ---------------------------------------------------------------------------
// Pass 2: single-block mean of both min arrays.
// ---------------------------------------------------------------------------
__global__ __launch_bounds__(256) void chamfer_finalize(
    const float* __restrict__ rmC, const float* __restrict__ rmT,
    float* __restrict__ out, int N, int M) {
  __shared__ float s[256];
  float sum = 0.0f;
  const float invN = 1.0f / (float)N, invM = 1.0f / (float)M;
  for (int i = threadIdx.x; i < N; i += 256) sum += rmC[i] * invN;
  for (int i = threadIdx.x; i < M; i += 256) sum += rmT[i] * invM;
  s[threadIdx.x] = sum;
  __syncthreads();
  for (int off = 128; off > 0; off >>= 1) {
    if ((int)threadIdx.x < off) s[threadIdx.x] += s[threadIdx.x + off];
    __syncthreads();
  }
  if (threadIdx.x == 0) out[0] = s[0];
}

extern "C" void kernel_launch(void* const* d_in, const int* in_sizes, int n_in,
                              void* d_out, int out_size, void* d_ws, size_t ws_size,
                              hipStream_t stream) {
  const float* Xc = (const float*)d_in[0];
  const float* Xt = (const float*)d_in[1];
  const int N = in_sizes[0] / 3;   // 16384
  const int M = in_sizes[1] / 3;   // 16384

  char* ws = (char*)d_ws;
  size_t off = 0;
  float4* Tc  = (float4*)(ws + off); off += (size_t)N * 16;
  float4* Tt  = (float4*)(ws + off); off += (size_t)M * 16;
  float2* BfC = (float2*)(ws + off); off += (size_t)N * 8;
  float2* BfT = (float2*)(ws + off); off += (size_t)M * 8;
  float*  rmC = (float*)(ws + off);  off += (size_t)N * 4;
  float*  rmT = (float*)(ws + off);

  const int nmax = (N > M) ? N : M;
  chamfer_pack_init<<<(nmax + 255) / 256, 256, 0, stream>>>(
      Xc, Xt, Tc, Tt, BfC, BfT, rmC, rmT, N, M);

  const int CHUNKS = 8;  // column chunks -> 64*8 blocks * 8 waves = 4096 waves/dir
  {
    dim3 grid(N / 256, CHUNKS);  // 8 waves/block * 32 rows/wave = 256 rows
    chamfer_min_kernel<<<grid, 256, 0, stream>>>(Tc, (const v2f*)BfT, rmC,
                                                 (M / 16) / CHUNKS);
  }
  {
    dim3 grid(M / 256, CHUNKS);
    chamfer_min_kernel<<<grid, 256, 0, stream>>>(Tt, (const v2f*)BfC, rmT,
                                                 (N / 16) / CHUNKS);
  }

  chamfer_finalize<<<1, 256, 0, stream>>>(rmC, rmT, (float*)d_out, N, M);
}